// HierarchicalPooling_38654705664490
// MI455X (gfx1250) — compile-verified
//
#include <hip/hip_runtime.h>
#include <math.h>

// Problem constants (fixed by the reference)
#define NN 400000
#define DD 256
#define GG 2048
#define D2 512     // 2*D
#define D5 1280    // 5*D
#define LDA 258    // padded LDS row stride (floats)

typedef __attribute__((ext_vector_type(2))) float v2f;
typedef __attribute__((ext_vector_type(8))) float v8f;

// ---- order-preserving float <-> uint key for atomicMax on floats ----
__device__ __forceinline__ unsigned f2key(float f) {
  unsigned b = __float_as_uint(f);
  return (b & 0x80000000u) ? ~b : (b | 0x80000000u);
}
__device__ __forceinline__ float key2f(unsigned u) {
  unsigned b = (u & 0x80000000u) ? (u ^ 0x80000000u) : ~u;
  return __uint_as_float(b);
}

// ---- gfx1250 async global->LDS DMA (ASYNCcnt-tracked), 16B per lane ----
// Generic pointers to __shared__ carry the wave-relative LDS offset in the
// low 32 bits.
__device__ __forceinline__ unsigned lds_off(const void* p) {
  return (unsigned)(size_t)p;
}
__device__ __forceinline__ void async_load_b128(unsigned lds_byte_off,
                                                const void* gaddr) {
  asm volatile("global_load_async_to_lds_b128 %0, %1, off"
               :: "v"(lds_byte_off), "v"(gaddr)
               : "memory");
}
__device__ __forceinline__ void wait_async0() {
  asm volatile("s_wait_asynccnt 0x0" ::: "memory");
}

// ---- one wave computes a 16x16 fp32 tile: C += A[16xK] * B[Kx16] ----
// A row-major with stride lda, B row-major with stride ldb.
// fp32 WMMA layout (ISA 7.12.2): lanes 0-15 hold K=k,k+1; lanes 16-31 K=k+2,k+3.
__device__ __forceinline__ v8f wmma_tile_f32(const float* A, int lda,
                                             const float* B, int ldb,
                                             int K, v8f c) {
  const int lane = threadIdx.x & 31;
  const int half = lane >> 4;
  const int l16  = lane & 15;
  const float* arow = A + (size_t)l16 * lda + 2 * half;
  #pragma unroll 4
  for (int k = 0; k < K; k += 4) {
    v2f a, b;
    a.x = arow[k];
    a.y = arow[k + 1];
    b.x = B[(size_t)(k + 2 * half) * ldb + l16];
    b.y = B[(size_t)(k + 2 * half + 1) * ldb + l16];
    c = __builtin_amdgcn_wmma_f32_16x16x4_f32(false, a, false, b, (short)0, c,
                                              false, false);
  }
  return c;
}

// ---- kernel 0: zero init of accumulator region (keys: 0 == minimal key) ----
__global__ void k0_zero(unsigned* p, long long n) {
  long long i = (long long)blockIdx.x * 256 + threadIdx.x;
  if (i < n) p[i] = 0u;
}

// ---- kernel 1: segment sum/max/count + attention scores (+ segment max of scores)
__global__ void k1_stats(const float* __restrict__ x, const int* __restrict__ batch,
                         const float* __restrict__ w_attn, const float* __restrict__ b_attn,
                         float* h_sum, unsigned* h_max_key, float* counts,
                         unsigned* m_attn_key, float* scores) {
  __shared__ int   bsh[128];
  __shared__ float wsh[DD];
  const int tid  = threadIdx.x;
  const int base = blockIdx.x * 128;
  if (tid < 128) { int n = base + tid; bsh[tid] = (n < NN) ? batch[n] : -1; }
  wsh[tid] = w_attn[tid];
  __syncthreads();

  // phase A: per-node dot(x, w_attn): one wave per node, 8 feats/lane + shuffle reduce
  const int w = tid >> 5, lane = tid & 31;
  for (int i = w; i < 128; i += 8) {
    int n = base + i;
    if (n >= NN) break;
    float p = 0.f;
    #pragma unroll
    for (int j = 0; j < 8; ++j)
      p += x[(size_t)n * DD + lane + 32 * j] * wsh[lane + 32 * j];
    for (int s = 16; s > 0; s >>= 1) p += __shfl_down(p, s);
    if (lane == 0) {
      float sc = p + b_attn[0];
      scores[n] = sc;
      atomicMax(&m_attn_key[bsh[i]], f2key(sc));
    }
  }

  // phase B: thread-per-feature running sum/max over sorted runs
  const int c = tid;
  float sacc = 0.f, macc = -3.4e38f;
  int cur = -1, rlen = 0;
  for (int i = 0; i < 128; ++i) {
    int n = base + i;
    if (n >= NN) break;
    int g = bsh[i];
    if (g != cur) {
      if (cur >= 0) {
        atomicAdd(&h_sum[(size_t)cur * DD + c], sacc);
        atomicMax(&h_max_key[(size_t)cur * DD + c], f2key(macc));
        if (c == 0) atomicAdd(&counts[cur], (float)rlen);
      }
      cur = g; sacc = 0.f; macc = -3.4e38f; rlen = 0;
    }
    float v = x[(size_t)n * DD + c];
    sacc += v; macc = fmaxf(macc, v); ++rlen;
  }
  if (cur >= 0) {
    atomicAdd(&h_sum[(size_t)cur * DD + c], sacc);
    atomicMax(&h_max_key[(size_t)cur * DD + c], f2key(macc));
    if (c == 0) atomicAdd(&counts[cur], (float)rlen);
  }
}

// ---- kernel 2: h_mean/h_max/h_sum into combined slots 0..2; q = h_mean@Wq + bq ----
__global__ void k2_meanmax_q(const float* __restrict__ h_sum, const unsigned* __restrict__ h_max_key,
                             const float* __restrict__ counts,
                             const float* __restrict__ Wq, const float* __restrict__ bq,
                             float* combined, float* q) {
  __shared__ float hm[DD];
  const int g = blockIdx.x, c = threadIdx.x;
  float cnt  = counts[g];
  float s    = h_sum[(size_t)g * DD + c];
  float mean = s / cnt;
  combined[(size_t)g * D5 + c]          = mean;
  combined[(size_t)g * D5 + DD + c]     = key2f(h_max_key[(size_t)g * DD + c]);
  combined[(size_t)g * D5 + 2 * DD + c] = s;
  hm[c] = mean;
  __syncthreads();
  float acc = bq[c];
  for (int r = 0; r < DD; ++r) acc += hm[r] * Wq[r * DD + c];
  q[(size_t)g * DD + c] = acc;
}

// ---- kernel 3: k = x@Wk + bk via WMMA (32-node tile), logits = q.k/16, segment max ----
__global__ void k3_logits(const float* __restrict__ x, const int* __restrict__ batch,
                          const float* __restrict__ Wk, const float* __restrict__ bk,
                          const float* __restrict__ q,
                          float* logits, unsigned* m_s2s_key) {
  __shared__ float xs[32 * LDA];
  __shared__ int   bsh[32];
  __shared__ float lpart[32];
  const int tid  = threadIdx.x;
  const int base = blockIdx.x * 32;           // NN % 32 == 0
  // async-DMA the 32x256 x tile into LDS (2048 x b128)
  for (int idx = tid; idx < 32 * DD / 4; idx += 256) {
    int r  = idx >> 6;            // 64 b128 transfers per row
    int c4 = (idx & 63) * 4;
    async_load_b128(lds_off(&xs[r * LDA + c4]),
                    &x[(size_t)(base + r) * DD + c4]);
  }
  if (tid < 32) { bsh[tid] = batch[base + tid]; lpart[tid] = 0.f; }
  wait_async0();
  __syncthreads();

  const int w = tid >> 5, lane = tid & 31, half = lane >> 4, l16 = lane & 15;
  for (int t = 0; t < 4; ++t) {
    int job = w + 8 * t;                      // 32 (16x16) tiles: 2 row-groups x 16 col-tiles
    int rg = job >> 4, ct = job & 15;
    v8f c = {};
    c = wmma_tile_f32(&xs[rg * 16 * LDA], LDA, Wk + ct * 16, DD, DD, c);
    #pragma unroll
    for (int j = 0; j < 8; ++j) {
      int row = rg * 16 + j + 8 * half;       // C layout: VGPR j -> M=j (lanes<16), M=j+8
      int col = ct * 16 + l16;
      float kv = c[j] + bk[col];
      atomicAdd(&lpart[row], kv * q[(size_t)bsh[row] * DD + col]);
    }
  }
  __syncthreads();
  if (tid < 32) {
    float lg = lpart[tid] * 0.0625f;          // 1/sqrt(256)
    logits[base + tid] = lg;
    atomicMax(&m_s2s_key[bsh[tid]], f2key(lg));
  }
}

// ---- kernel 4: e = exp(v - segmax), accumulate segment Z (both softmaxes) ----
__global__ void k4_exp(const int* __restrict__ batch,
                       const unsigned* __restrict__ m_attn_key, const unsigned* __restrict__ m_s2s_key,
                       float* scores, float* logits, float* z_attn, float* z_s2s) {
  int n = blockIdx.x * 256 + threadIdx.x;
  if (n >= NN) return;
  int g = batch[n];
  float e1 = expf(scores[n] - key2f(m_attn_key[g]));
  scores[n] = e1;
  atomicAdd(&z_attn[g], e1);
  float e2 = expf(logits[n] - key2f(m_s2s_key[g]));
  logits[n] = e2;
  atomicAdd(&z_s2s[g], e2);
}

// ---- kernel 5: recompute k (WMMA, 16-node tile); accumulate h_attn and h_s2s ----
__global__ void k5_weighted(const float* __restrict__ x, const int* __restrict__ batch,
                            const float* __restrict__ Wk, const float* __restrict__ bk,
                            const float* __restrict__ scores_e, const float* __restrict__ logits_e,
                            const float* __restrict__ z_attn, const float* __restrict__ z_s2s,
                            float* combined) {
  __shared__ float xs[16 * LDA];
  __shared__ float ks[16 * LDA];
  __shared__ float aw[16], ww[16];
  __shared__ int   bsh[16];
  const int tid  = threadIdx.x;
  const int base = blockIdx.x * 16;           // NN % 16 == 0
  for (int idx = tid; idx < 16 * DD / 4; idx += 256) {
    int r  = idx >> 6;
    int c4 = (idx & 63) * 4;
    async_load_b128(lds_off(&xs[r * LDA + c4]),
                    &x[(size_t)(base + r) * DD + c4]);
  }
  if (tid < 16) {
    int n = base + tid;
    int g = batch[n];
    bsh[tid] = g;
    aw[tid]  = scores_e[n] / z_attn[g];
    ww[tid]  = logits_e[n] / z_s2s[g];
  }
  wait_async0();
  __syncthreads();

  const int w = tid >> 5, lane = tid & 31, half = lane >> 4, l16 = lane & 15;
  for (int t = 0; t < 2; ++t) {
    int ct = w + 8 * t;                       // 16 col-tiles
    v8f c = {};
    c = wmma_tile_f32(xs, LDA, Wk + ct * 16, DD, DD, c);
    #pragma unroll
    for (int j = 0; j < 8; ++j) {
      int row = j + 8 * half;
      int col = ct * 16 + l16;
      ks[row * LDA + col] = c[j] + bk[col];
    }
  }
  __syncthreads();

  const int c = tid;
  float aacc = 0.f, sacc = 0.f;
  int cur = -1;
  for (int i = 0; i < 16; ++i) {
    int g = bsh[i];
    if (g != cur) {
      if (cur >= 0) {
        atomicAdd(&combined[(size_t)cur * D5 + 3 * DD + c], aacc);
        atomicAdd(&combined[(size_t)cur * D5 + 4 * DD + c], sacc);
      }
      cur = g; aacc = 0.f; sacc = 0.f;
    }
    aacc += xs[i * LDA + c] * aw[i];
    sacc += ks[i * LDA + c] * ww[i];
  }
  if (cur >= 0) {
    atomicAdd(&combined[(size_t)cur * D5 + 3 * DD + c], aacc);
    atomicAdd(&combined[(size_t)cur * D5 + 4 * DD + c], sacc);
  }
}

// ---- WMMA GEMM for the MLP: C = act(A[MxK] @ B[KxNc] + bias) ----
// 8 waves/block, 1 16x16 tile/wave; all 8 tiles of a block share the same
// 16-row A stripe (8 | Nc/16), so the stripe is async-staged in LDS in
// 256-wide K chunks.
template <int GELU>
__global__ void k_gemm(const float* __restrict__ A, const float* __restrict__ B,
                       const float* __restrict__ bias, float* __restrict__ C,
                       int M, int K, int Nc) {
  __shared__ float as_[16 * LDA];
  const int tid  = threadIdx.x;
  const int tile = blockIdx.x * 8 + (tid >> 5);
  const int ntx  = Nc >> 4;
  const int tm = tile / ntx, tn = tile - tm * ntx;   // tm uniform across block
  v8f c = {};
  for (int k0 = 0; k0 < K; k0 += 256) {
    for (int idx = tid; idx < 16 * 256 / 4; idx += 256) {
      int r  = idx >> 6;
      int c4 = (idx & 63) * 4;
      async_load_b128(lds_off(&as_[r * LDA + c4]),
                      &A[(size_t)(tm * 16 + r) * K + k0 + c4]);
    }
    wait_async0();
    __syncthreads();
    c = wmma_tile_f32(as_, LDA, B + (size_t)k0 * Nc + tn * 16, Nc, 256, c);
    __syncthreads();
  }
  const int lane = tid & 31, half = lane >> 4, l16 = lane & 15;
  #pragma unroll
  for (int j = 0; j < 8; ++j) {
    int row = tm * 16 + j + 8 * half;
    int col = tn * 16 + l16;
    float v = c[j] + bias[col];
    if (GELU) v = 0.5f * v * (1.f + erff(v * 0.70710678118654752f));
    C[(size_t)row * Nc + col] = v;
  }
}

extern "C" void kernel_launch(void* const* d_in, const int* in_sizes, int n_in,
                              void* d_out, int out_size, void* d_ws, size_t ws_size,
                              hipStream_t stream) {
  (void)in_sizes; (void)n_in; (void)out_size; (void)ws_size;
  const float* x      = (const float*)d_in[0];
  const int*   batch  = (const int*)d_in[1];
  // d_in[2] = n_graphs (compile-time constant GG)
  const float* w_attn = (const float*)d_in[3];
  const float* b_attn = (const float*)d_in[4];
  const float* Wq     = (const float*)d_in[5];
  const float* bq     = (const float*)d_in[6];
  const float* Wk     = (const float*)d_in[7];
  const float* bk     = (const float*)d_in[8];
  const float* W1     = (const float*)d_in[9];
  const float* b1     = (const float*)d_in[10];
  const float* W2     = (const float*)d_in[11];
  const float* b2     = (const float*)d_in[12];

  // workspace layout (floats); zero-init region first, then scratch
  float*    ws       = (float*)d_ws;
  float*    h_sum    = ws;                                   // G*D
  unsigned* h_maxk   = (unsigned*)(ws + (size_t)GG * DD);    // G*D
  float*    counts   = ws + (size_t)2 * GG * DD;             // G
  unsigned* m_ak     = (unsigned*)(counts + GG);             // G
  float*    z_attn   = counts + 2 * GG;                      // G
  unsigned* m_sk     = (unsigned*)(counts + 3 * GG);         // G
  float*    z_s2s    = counts + 4 * GG;                      // G
  float*    combined = counts + 5 * GG;                      // G*5D
  float*    q        = combined + (size_t)GG * D5;           // G*D
  float*    scores   = q + (size_t)GG * DD;                  // N
  float*    logits   = scores + NN;                          // N
  float*    hmid     = logits + NN;                          // G*2D

  long long zn = (long long)2 * GG * DD + 5 * GG + (long long)GG * D5;
  k0_zero<<<(unsigned)((zn + 255) / 256), 256, 0, stream>>>((unsigned*)ws, zn);

  k1_stats<<<(NN + 127) / 128, 256, 0, stream>>>(x, batch, w_attn, b_attn,
                                                 h_sum, h_maxk, counts, m_ak, scores);
  k2_meanmax_q<<<GG, 256, 0, stream>>>(h_sum, h_maxk, counts, Wq, bq, combined, q);
  k3_logits<<<NN / 32, 256, 0, stream>>>(x, batch, Wk, bk, q, logits, m_sk);
  k4_exp<<<(NN + 255) / 256, 256, 0, stream>>>(batch, m_ak, m_sk, scores, logits,
                                               z_attn, z_s2s);
  k5_weighted<<<NN / 16, 256, 0, stream>>>(x, batch, Wk, bk, scores, logits,
                                           z_attn, z_s2s, combined);
  k_gemm<1><<<(GG / 16) * (D2 / 16) / 8, 256, 0, stream>>>(combined, W1, b1, hmid,
                                                           GG, D5, D2);
  k_gemm<0><<<(GG / 16) * (DD / 16) / 8, 256, 0, stream>>>(hmid, W2, b2,
                                                           (float*)d_out, GG, D2, DD);
}